// SimpleGNN_74981539053649
// MI455X (gfx1250) — compile-verified
//
#include <hip/hip_runtime.h>

#define N_IN  128
#define N_HID 64
#define N_OUT 16

typedef float v2f __attribute__((ext_vector_type(2)));
typedef float v8f __attribute__((ext_vector_type(8)));

// ---------------- utility kernels ----------------

__global__ __launch_bounds__(256) void fill_kernel(float* __restrict__ p, float v, long long n) {
  long long i = blockIdx.x * 256LL + threadIdx.x;
  if (i < n) p[i] = v;
}

// deg[dst] += 1 for every edge (deg pre-filled with 1.0 for the self-loop)
__global__ __launch_bounds__(256) void degree_kernel(const int* __restrict__ dst,
                                                     float* __restrict__ deg, long long E) {
  long long i = blockIdx.x * 256LL + threadIdx.x;
  if (i < E) atomicAdd(&deg[dst[i]], 1.0f);
}

__global__ __launch_bounds__(256) void rsqrt_kernel(float* __restrict__ d, int n) {
  int i = blockIdx.x * 256 + threadIdx.x;
  if (i < n) {
    float v = d[i];
    d[i] = rsqrtf(v < 1.0f ? 1.0f : v);
  }
}

// ---------------- GEMM1: H[N,64] = X[N,128] @ W[128,64], f32 WMMA ----------------
// Block = 128 threads = 4 waves, each wave computes a 16x64 row-slab.
__global__ __launch_bounds__(128) void gemm1_wmma(const float* __restrict__ X,
                                                  const float* __restrict__ W,
                                                  float* __restrict__ H, int nRows) {
  __shared__ float Wt[N_HID * N_IN];  // W transposed: Wt[n][k], 32 KB
  for (int i = threadIdx.x; i < N_IN * N_HID; i += 128) {
    int k = i / N_HID, n = i % N_HID;
    Wt[n * N_IN + k] = W[i];
  }
  __syncthreads();

  const int wave   = threadIdx.x >> 5;
  const int lane   = threadIdx.x & 31;
  const int laneLo = lane & 15;
  const int laneHi = lane >> 4;  // 0: K pair {0,1}, 1: K pair {2,3}
  const int rowBase = (blockIdx.x * 4 + wave) * 16;

  // clamp row for loads so EXEC stays all-ones across the WMMA (ISA §7.12 requirement)
  int rowA = rowBase + laneLo;
  if (rowA >= nRows) rowA = nRows - 1;
  const float* xrow = X + (size_t)rowA * N_IN;

  v8f acc[4] = {};
  for (int kb = 0; kb < N_IN; kb += 4) {
    // A 16x4 fragment: float2 per lane (K pair selected by lane half)
    v2f a = *(const v2f*)(xrow + kb + 2 * laneHi);
#pragma unroll
    for (int t = 0; t < 4; ++t) {
      // B 4x16 fragment from transposed-W LDS: contiguous K pair per lane
      v2f b = *(const v2f*)(&Wt[(t * 16 + laneLo) * N_IN + kb + 2 * laneHi]);
      acc[t] = __builtin_amdgcn_wmma_f32_16x16x4_f32(
          /*neg_a=*/false, a, /*neg_b=*/false, b,
          /*c_mod=*/(short)0, acc[t], /*reuse_a=*/false, /*reuse_b=*/false);
    }
  }

  // C/D layout: VGPR v -> row rowBase+v+laneHi*8, col = t*16 + laneLo
#pragma unroll
  for (int t = 0; t < 4; ++t) {
    int col = t * 16 + laneLo;
#pragma unroll
    for (int v = 0; v < 8; ++v) {
      int r = rowBase + v + laneHi * 8;
      if (r < nRows) H[(size_t)r * N_HID + col] = acc[t][v];
    }
  }
}

// ---------------- GEMM2: H2[N,16] = Hrelu[N,64] @ W2[64,16], f32 WMMA ----------------
__global__ __launch_bounds__(128) void gemm2_wmma(const float* __restrict__ Hrelu,
                                                  const float* __restrict__ W,
                                                  float* __restrict__ H2, int nRows) {
  __shared__ float Wt[N_OUT * N_HID];  // Wt[n][k], 4 KB
  for (int i = threadIdx.x; i < N_HID * N_OUT; i += 128) {
    int k = i / N_OUT, n = i % N_OUT;
    Wt[n * N_HID + k] = W[i];
  }
  __syncthreads();

  const int wave   = threadIdx.x >> 5;
  const int lane   = threadIdx.x & 31;
  const int laneLo = lane & 15;
  const int laneHi = lane >> 4;
  const int rowBase = (blockIdx.x * 4 + wave) * 16;

  int rowA = rowBase + laneLo;
  if (rowA >= nRows) rowA = nRows - 1;
  const float* hrow = Hrelu + (size_t)rowA * N_HID;

  v8f acc = {};
  for (int kb = 0; kb < N_HID; kb += 4) {
    v2f a = *(const v2f*)(hrow + kb + 2 * laneHi);
    v2f b = *(const v2f*)(&Wt[laneLo * N_HID + kb + 2 * laneHi]);
    acc = __builtin_amdgcn_wmma_f32_16x16x4_f32(false, a, false, b, (short)0, acc, false, false);
  }

#pragma unroll
  for (int v = 0; v < 8; ++v) {
    int r = rowBase + v + laneHi * 8;
    if (r < nRows) H2[(size_t)r * N_OUT + laneLo] = acc[v];
  }
}

// ---------------- edge scatter: agg[dst,f] += h[src,f] * dinv[src]*dinv[dst] ----------------
template <int F, int SHIFT>
__global__ __launch_bounds__(256) void scatter_kernel(const int* __restrict__ src,
                                                      const int* __restrict__ dst,
                                                      const float* __restrict__ dinv,
                                                      const float* __restrict__ h,
                                                      float* __restrict__ agg,
                                                      long long total) {
  long long idx = blockIdx.x * 256LL + threadIdx.x;
  if (idx >= total) return;
  int       f = (int)(idx & (F - 1));
  long long e = idx >> SHIFT;
  int s = src[e];
  int d = dst[e];
  float norm = dinv[s] * dinv[d];
  atomicAdd(&agg[(size_t)d * F + f], h[(size_t)s * F + f] * norm);
}

// ---------------- self-loop + bias + relu (layer 1, in place on agg1) ----------------
__global__ __launch_bounds__(256) void finalize1_kernel(float* __restrict__ agg1,
                                                        const float* __restrict__ h1,
                                                        const float* __restrict__ dinv,
                                                        const float* __restrict__ b, int n) {
  long long idx = blockIdx.x * 256LL + threadIdx.x;
  if (idx >= (long long)n * N_HID) return;
  int i = (int)(idx >> 6);
  int f = (int)(idx & 63);
  float di = dinv[i];
  float v = agg1[idx] + h1[idx] * di * di + b[f];
  agg1[idx] = v > 0.0f ? v : 0.0f;
}

// ---------------- self-loop + bias (layer 2) -> node_emb ----------------
__global__ __launch_bounds__(256) void finalize2_kernel(const float* __restrict__ agg2,
                                                        const float* __restrict__ h2,
                                                        const float* __restrict__ dinv,
                                                        const float* __restrict__ b,
                                                        float* __restrict__ out, int n) {
  long long idx = blockIdx.x * 256LL + threadIdx.x;
  if (idx >= (long long)n * N_OUT) return;
  int i = (int)(idx >> 4);
  int f = (int)(idx & 15);
  float di = dinv[i];
  out[idx] = agg2[idx] + h2[idx] * di * di + b[f];
}

// ---------------- mean pool (single graph): LDS partial sums, one global atomic per block ----------------
__global__ __launch_bounds__(256) void pool_kernel(const float* __restrict__ emb,
                                                   float* __restrict__ pool, long long total) {
  __shared__ float s[N_OUT];
  if (threadIdx.x < N_OUT) s[threadIdx.x] = 0.0f;
  __syncthreads();
  long long stride = (long long)gridDim.x * 256;
  for (long long idx = blockIdx.x * 256LL + threadIdx.x; idx < total; idx += stride)
    atomicAdd(&s[(int)(idx & 15)], emb[idx]);
  __syncthreads();
  if (threadIdx.x < N_OUT) atomicAdd(&pool[threadIdx.x], s[threadIdx.x]);
}

__global__ void pooldiv_kernel(const float* __restrict__ pool, float* __restrict__ gout, float cnt) {
  int f = threadIdx.x;
  if (f < N_OUT) gout[f] = pool[f] / (cnt < 1.0f ? 1.0f : cnt);
}

// ---------------- host launch ----------------
extern "C" void kernel_launch(void* const* d_in, const int* in_sizes, int n_in,
                              void* d_out, int out_size, void* d_ws, size_t ws_size,
                              hipStream_t stream) {
  (void)n_in; (void)out_size; (void)ws_size;
  const float* x  = (const float*)d_in[0];
  const int*   ei = (const int*)d_in[1];
  // d_in[2] = batch (all zeros, single graph) -> mean over all nodes
  const float* W1 = (const float*)d_in[3];
  const float* b1 = (const float*)d_in[4];
  const float* W2 = (const float*)d_in[5];
  const float* b2 = (const float*)d_in[6];
  float* out = (float*)d_out;

  const int       N = in_sizes[0] / N_IN;        // 100000
  const long long E = (long long)in_sizes[1] / 2; // 1600000
  const int* srcIdx = ei;
  const int* dstIdx = ei + E;

  // workspace layout (bytes):
  //   [0, 4N)           dinv (deg -> rsqrt in place)
  //   [512K, 512K+64)   pool accumulators
  //   [1M, 1M+256N)     h1 [N,64]; after finalize1, reused: h2 [N,16] + agg2 [N,16]
  //   [1M+256N, ...)    agg1 [N,64] (also holds relu output)
  char*  ws   = (char*)d_ws;
  float* dinv = (float*)ws;
  float* pool = (float*)(ws + (512 << 10));
  float* h1   = (float*)(ws + (1 << 20));
  float* agg1 = h1 + (size_t)N * N_HID;
  float* h2   = h1;                         // overlays dead h1
  float* agg2 = h1 + (size_t)N * N_OUT;     // overlays dead h1 (disjoint from h2)

  auto nblk = [](long long n, int t) { return (unsigned)((n + t - 1) / t); };

  // degree + normalization
  fill_kernel<<<nblk(N, 256), 256, 0, stream>>>(dinv, 1.0f, N);               // self-loop
  fill_kernel<<<nblk((long long)N * N_HID, 256), 256, 0, stream>>>(agg1, 0.0f, (long long)N * N_HID);
  fill_kernel<<<1, 256, 0, stream>>>(pool, 0.0f, N_OUT);
  degree_kernel<<<nblk(E, 256), 256, 0, stream>>>(dstIdx, dinv, E);
  rsqrt_kernel<<<nblk(N, 256), 256, 0, stream>>>(dinv, N);

  // layer 1
  gemm1_wmma<<<nblk(N, 64), 128, 0, stream>>>(x, W1, h1, N);
  scatter_kernel<N_HID, 6><<<nblk(E * N_HID, 256), 256, 0, stream>>>(
      srcIdx, dstIdx, dinv, h1, agg1, E * N_HID);
  finalize1_kernel<<<nblk((long long)N * N_HID, 256), 256, 0, stream>>>(agg1, h1, dinv, b1, N);

  // layer 2 (h1 region now dead; zero agg2 AFTER finalize1 because it overlays h1)
  fill_kernel<<<nblk((long long)N * N_OUT, 256), 256, 0, stream>>>(agg2, 0.0f, (long long)N * N_OUT);
  gemm2_wmma<<<nblk(N, 64), 128, 0, stream>>>(agg1, W2, h2, N);
  scatter_kernel<N_OUT, 4><<<nblk(E * N_OUT, 256), 256, 0, stream>>>(
      srcIdx, dstIdx, dinv, h2, agg2, E * N_OUT);
  finalize2_kernel<<<nblk((long long)N * N_OUT, 256), 256, 0, stream>>>(agg2, h2, dinv, b2, out, N);

  // graph mean pool -> out[N*16 .. N*16+15]
  pool_kernel<<<512, 256, 0, stream>>>(out, pool, (long long)N * N_OUT);
  pooldiv_kernel<<<1, 32, 0, stream>>>(pool, out + (size_t)N * N_OUT, (float)N);
}